// TMN4_GodView_AGI_26688926777894
// MI455X (gfx1250) — compile-verified
//
#include <hip/hip_runtime.h>
#include <hip/hip_bf16.h>

#define DIM   512
#define TOPK  12
#define NTOK  64

typedef __attribute__((ext_vector_type(2))) float v2f;
typedef __attribute__((ext_vector_type(4))) float v4f;
typedef __attribute__((ext_vector_type(8))) float v8f;

// ---------------------------------------------------------------------------
// Kernel 1: encode — vec[j] = mean_t (sym+vis+phys)[tok[t], j] / 3
// ---------------------------------------------------------------------------
__global__ void encode_kernel(const float* __restrict__ sym,
                              const float* __restrict__ vis,
                              const float* __restrict__ phys,
                              const int*   __restrict__ tokens,
                              float* __restrict__ vec) {
    int j = threadIdx.x;                       // 512 threads
    float acc = 0.0f;
    for (int t = 0; t < NTOK; ++t) {
        size_t off = (size_t)tokens[t] * DIM + j;   // tokens[t] is wave-uniform -> SMEM
        acc += sym[off] + vis[off] + phys[off];
    }
    vec[j] = acc / (3.0f * (float)NTOK);
}

// ---------------------------------------------------------------------------
// Kernel 2: fused 3-table weighted matvec via V_WMMA_F32_16X16X4_F32.
// sim[v] = 0.5*dot(sym[v],vec) + 0.3*dot(vis[v],vec) + 0.2*dot(phys[v],vec)
//
// Per wave: 16 rows. A tile (16x4 f32): lanes 0-15 hold k-slots {0,1},
// lanes 16-31 hold k-slots {2,3}. Dot products are K-permutation invariant:
// lane half 0 loads row[k..k+3] (feeding slots 0,1 of two WMMAs), lane half 1
// loads row[k+4..k+7] (feeding slots 2,3). B broadcasts the matching
// pre-scaled vec entries (identical across all 16 N lanes, from LDS), so all
// columns of D equal the 16 row-dots. The three tables accumulate into one
// f32 accumulator using 0.5/0.3/0.2 pre-scaled copies of vec in LDS.
//
// Table reads are non-temporal: 614 MB/pass streamed 3x >> 192 MB L2, so
// cyclic RT caching gets ~0% reuse and only thrashes; NT is the right hint.
// Taking the kernel-arg pointer directly (no pointer array) keeps the loads
// in the global address space -> global_load_b128, not flat_load_b128.
// ---------------------------------------------------------------------------
__device__ __forceinline__ v8f accum_table(const float* __restrict__ T,
                                           const float* __restrict__ wv,
                                           int half, v8f acc) {
    #pragma unroll 4
    for (int k0 = 0; k0 < DIM; k0 += 8) {
        // global b128 (NT): half 0 -> row[k0..k0+3], half 1 -> row[k0+4..k0+7]
        v4f a = __builtin_nontemporal_load((const v4f*)(T + k0 + half * 4));
        // LDS b128: matching pre-scaled vec entries for this lane-half
        v4f b = *(const v4f*)(wv + k0 + half * 4);

        v2f A1 = { a.x, a.y };  v2f B1 = { b.x, b.y };
        v2f A2 = { a.z, a.w };  v2f B2 = { b.z, b.w };

        acc = __builtin_amdgcn_wmma_f32_16x16x4_f32(
                  false, A1, false, B1, (short)0, acc, false, false);
        acc = __builtin_amdgcn_wmma_f32_16x16x4_f32(
                  false, A2, false, B2, (short)0, acc, false, false);
    }
    return acc;
}

__global__ __launch_bounds__(256)
void sim_wmma_kernel(const float* __restrict__ sym,
                     const float* __restrict__ vis,
                     const float* __restrict__ phys,
                     const float* __restrict__ vec,
                     float* __restrict__ out, int V) {
    __shared__ __align__(16) float svec[3 * DIM];   // 0.5*vec | 0.3*vec | 0.2*vec

    int tid = threadIdx.x;
    for (int i = tid; i < DIM; i += blockDim.x) {
        float v = vec[i];
        svec[i]           = 0.5f * v;
        svec[DIM + i]     = 0.3f * v;
        svec[2 * DIM + i] = 0.2f * v;
    }
    __syncthreads();

    int wave = tid >> 5;        // 8 waves per block
    int lane = tid & 31;
    int half = lane >> 4;       // 0: k-slots {0,1}, 1: k-slots {2,3}
    int l16  = lane & 15;

    int rowBase = (blockIdx.x * 8 + wave) * 16;     // 16 rows per wave
    if (rowBase >= V) return;                        // wave-uniform exit

    int row  = rowBase + l16;
    int rowc = row < V ? row : (V - 1);              // tail clamp (no divergence)
    size_t rowOff = (size_t)rowc * DIM;

    v8f acc = {};                                    // 16x16 f32 accumulator
    acc = accum_table(sym  + rowOff, &svec[0],       half, acc);
    acc = accum_table(vis  + rowOff, &svec[DIM],     half, acc);
    acc = accum_table(phys + rowOff, &svec[2 * DIM], half, acc);

    // D VGPR i: lanes 0-15 = M=i, lanes 16-31 = M=8+i (all N identical).
    // Lane (half, l16<8) emits row rowBase + half*8 + l16 from acc[l16].
    int q = l16 & 7;
    float r = acc[0];
    #pragma unroll
    for (int i = 1; i < 8; ++i) if (q == i) r = acc[i];

    int orow = rowBase + half * 8 + l16;
    if (l16 < 8 && orow < V) out[orow] = r;
}

// ---------------------------------------------------------------------------
// Kernel 3: top-12 of sim[V]. 256 threads: register-sorted per-thread top-12
// lists, then LDS tree merge (24 KB).
// ---------------------------------------------------------------------------
__global__ void topk_kernel(const float* __restrict__ sim, int V,
                            float* __restrict__ tv, int* __restrict__ ti) {
    __shared__ float sval[256 * TOPK];
    __shared__ int   sidx[256 * TOPK];
    int t = threadIdx.x;

    float v[TOPK]; int id[TOPK];
    #pragma unroll
    for (int i = 0; i < TOPK; ++i) { v[i] = -3.0e38f; id[i] = 0; }

    for (int x = t; x < V; x += 256) {
        float s = sim[x];
        if (s > v[TOPK - 1]) {
            v[TOPK - 1] = s; id[TOPK - 1] = x;
            #pragma unroll
            for (int i = TOPK - 1; i > 0; --i) {
                if (v[i] > v[i - 1]) {
                    float fv = v[i]; v[i] = v[i - 1]; v[i - 1] = fv;
                    int   iv = id[i]; id[i] = id[i - 1]; id[i - 1] = iv;
                }
            }
        }
    }
    #pragma unroll
    for (int i = 0; i < TOPK; ++i) { sval[t * TOPK + i] = v[i]; sidx[t * TOPK + i] = id[i]; }
    __syncthreads();

    for (int st = 128; st >= 1; st >>= 1) {
        if (t < st) {
            float* a  = &sval[t * TOPK];         int* ai = &sidx[t * TOPK];
            float* b  = &sval[(t + st) * TOPK];  int* bi = &sidx[(t + st) * TOPK];
            float mv[TOPK]; int mi[TOPK];
            int pa = 0, pb = 0;
            #pragma unroll
            for (int i = 0; i < TOPK; ++i) {      // pa+pb == i <= 11 at read time
                if (a[pa] >= b[pb]) { mv[i] = a[pa]; mi[i] = ai[pa]; ++pa; }
                else               { mv[i] = b[pb]; mi[i] = bi[pb]; ++pb; }
            }
            #pragma unroll
            for (int i = 0; i < TOPK; ++i) { a[i] = mv[i]; ai[i] = mi[i]; }
        }
        __syncthreads();
    }
    if (t < TOPK) { tv[t] = sval[t]; ti[t] = sidx[t]; }
}

// ---------------------------------------------------------------------------
// Kernel 4: vec[j] = sum_i tv[i]*(sym+vis+phys)[ti[i], j] / 3
// ---------------------------------------------------------------------------
__global__ void gather_kernel(const float* __restrict__ sym,
                              const float* __restrict__ vis,
                              const float* __restrict__ phys,
                              const float* __restrict__ tv,
                              const int*   __restrict__ ti,
                              float* __restrict__ vec) {
    int j = threadIdx.x;                        // 512 threads
    float acc = 0.0f;
    #pragma unroll
    for (int i = 0; i < TOPK; ++i) {
        float w = tv[i];
        size_t off = (size_t)ti[i] * DIM + j;
        acc += w * (sym[off] + vis[off] + phys[off]);
    }
    vec[j] = acc * (1.0f / 3.0f);
}

// ---------------------------------------------------------------------------
// Kernel 5: ctx = (vec @ L0) @ L1 - 0.8 * normalize(self)
// ---------------------------------------------------------------------------
__global__ void stack_kernel(const float* __restrict__ vec,
                             const float* __restrict__ L0,
                             const float* __restrict__ L1,
                             const float* __restrict__ selfe,
                             float* __restrict__ ctx) {
    __shared__ float vin[DIM];
    __shared__ float tmp[DIM];
    __shared__ float red[DIM];
    int j = threadIdx.x;                        // 512 threads

    vin[j] = vec[j];
    float s = selfe[j];
    red[j] = s * s;
    __syncthreads();
    for (int st = DIM / 2; st > 0; st >>= 1) {
        if (j < st) red[j] += red[j + st];
        __syncthreads();
    }
    float n1 = sqrtf(red[0]);
    // reference: u = self/n1 (if n1>1e-8), then ||u||==1 > 0.8 -> u *= 0.8
    float u = (n1 > 1e-8f) ? (0.8f * s / n1) : s;

    float a0 = 0.0f;
    for (int i = 0; i < DIM; ++i) a0 += vin[i] * L0[i * DIM + j];
    tmp[j] = a0;
    __syncthreads();

    float a1 = 0.0f;
    for (int i = 0; i < DIM; ++i) a1 += tmp[i] * L1[i * DIM + j];
    ctx[j] = a1 - u;
}

// ---------------------------------------------------------------------------
extern "C" void kernel_launch(void* const* d_in, const int* in_sizes, int n_in,
                              void* d_out, int out_size, void* d_ws, size_t ws_size,
                              hipStream_t stream) {
    const float* sym   = (const float*)d_in[0];
    const float* vis   = (const float*)d_in[1];
    const float* phys  = (const float*)d_in[2];
    const float* selfe = (const float*)d_in[3];
    const float* L0    = (const float*)d_in[4];
    const float* L1    = (const float*)d_in[5];
    const int*   tok   = (const int*)d_in[6];
    float* out = (float*)d_out;

    const int V = in_sizes[0] / DIM;            // 100000

    char* ws = (char*)d_ws;
    float* vec = (float*)(ws);                       // 512 f
    float* ctx = (float*)(ws + 2048);                // 512 f
    float* sim = (float*)(ws + 4096);                // V f
    float* tv  = (float*)(ws + 4096 + 400128);       // 12 f (padded)
    int*   ti  = (int*)  (ws + 4096 + 400128 + 64);  // 12 i

    int simBlocks = (V + 127) / 128;            // 128 rows per 256-thread block

    encode_kernel<<<1, DIM, 0, stream>>>(sym, vis, phys, tok, vec);
    for (int r = 0; r < 2; ++r) {
        sim_wmma_kernel<<<simBlocks, 256, 0, stream>>>(sym, vis, phys, vec, sim, V);
        topk_kernel<<<1, 256, 0, stream>>>(sim, V, tv, ti);
        gather_kernel<<<1, DIM, 0, stream>>>(sym, vis, phys, tv, ti, vec);
    }
    stack_kernel<<<1, DIM, 0, stream>>>(vec, L0, L1, selfe, ctx);
    sim_wmma_kernel<<<simBlocks, 256, 0, stream>>>(sym, vis, phys, ctx, out, V);
}